// edGNNLayer_82549271429644
// MI455X (gfx1250) — compile-verified
//
#include <hip/hip_runtime.h>

// ---------------------------------------------------------------------------
// edGNN layer for MI455X (gfx1250, wave32)
// Phase 0: zero agg mailboxes in workspace
// Phase 1: edge scatter with native GLOBAL_ATOMIC_ADD_F32 (inline asm,
//          non-returning -> STOREcnt only). This phase bounds the kernel:
//          ~0.6 GB of traffic -> ~26 us at 23.3 TB/s HBM.
// Phase 2: node update GEMM with V_WMMA_F32_16X16X4_F32 (exact fp32), fused
//          bias (as WMMA C input) + ReLU, nontemporal stores.
// ---------------------------------------------------------------------------

namespace {
constexpr int kNodes   = 50000;
constexpr int kEdges   = 800000;
constexpr int kNodeDim = 64;
constexpr int kEdgeDim = 32;
constexpr int kAggDim  = 96;                 // NODE_DIM + EDGE_DIM
constexpr int kInDim   = 160;                // NODE_DIM + kAggDim
constexpr int kOut     = 64;
constexpr int kRowTiles = kNodes / 16;       // 3125 (exact)
}

typedef __attribute__((ext_vector_type(2))) float v2f;
typedef __attribute__((ext_vector_type(8))) float v8f;

// Native fp32 global atomic add, non-returning (tracked on STOREcnt; the
// implicit wait-idle at s_endpgm guarantees completion before kernel end).
__device__ __forceinline__ void atomic_add_f32_native(float* p, float v) {
  asm volatile("global_atomic_add_f32 %0, %1, off"
               :
               : "v"(p), "v"(v)
               : "memory");
}

// ---------------------------------------------------------------- Phase 0 ---
__global__ void zero_ws_kernel(float4* __restrict__ p, int n4) {
  int i = blockIdx.x * blockDim.x + threadIdx.x;
  if (i < n4) p[i] = make_float4(0.f, 0.f, 0.f, 0.f);
}

// ---------------------------------------------------------------- Phase 1 ---
// One 32-lane slice per edge. Lane L handles mailbox features {L, L+32, L+64}:
//   L      -> nf[src][L]        (coalesced 128B gather)
//   L+32   -> nf[src][L+32]     (coalesced 128B gather)
//   L+64   -> ef[e][L]          (coalesced 128B stream)
// Atomic adds hit consecutive addresses within agg[dst] -> good L2 combining.
__global__ void edge_scatter_kernel(const float* __restrict__ nf,
                                    const float* __restrict__ ef,
                                    const int*  __restrict__ src,
                                    const int*  __restrict__ dst,
                                    float* __restrict__ agg) {
  unsigned t    = blockIdx.x * blockDim.x + threadIdx.x;
  unsigned e    = t >> 5;
  unsigned lane = t & 31u;
  if (e >= (unsigned)kEdges) return;

  int s = src[e];
  int d = dst[e];

  float v0 = nf[(size_t)s * kNodeDim + lane];
  float v1 = nf[(size_t)s * kNodeDim + lane + 32];
  float v2 = ef[(size_t)e * kEdgeDim + lane];

  float* row = agg + (size_t)d * kAggDim;
  atomic_add_f32_native(row + lane,      v0);
  atomic_add_f32_native(row + lane + 32, v1);
  atomic_add_f32_native(row + lane + 64, v2);
}

// ---------------------------------------------------------------- Phase 2 ---
// One wave per 16-node row tile; computes a 16x64 output tile as four 16x16
// WMMA accumulators. X = [nf | agg] (160 cols), B-frag = W^T tile.
//
// A 16x4 fp32 layout (ISA 7.12.2): lanes 0-15 hold {K=k,k+1}, lanes 16-31 hold
// {K=k+2,k+3}, M = lane&15.  C/D: VGPR r -> row (r + 8*(lane>>4)), col lane&15.
// K-loop is split at the nf|agg boundary so each half has a single base
// pointer (no per-iteration cndmask selects).
__global__ void node_update_wmma_kernel(const float* __restrict__ nf,
                                        const float* __restrict__ agg,
                                        const float* __restrict__ W,
                                        const float* __restrict__ b,
                                        float* __restrict__ out) {
  int wave = (int)((blockIdx.x * blockDim.x + threadIdx.x) >> 5);
  int lane = (int)(threadIdx.x & 31u);
  if (wave >= kRowTiles) return;          // wave-uniform: EXEC stays all-ones

  const int half    = lane >> 4;          // 0: K lo pair, 1: K hi pair
  const int lm      = lane & 15;
  const int rowbase = wave * 16;
  const int arow    = rowbase + lm;
  const int koff    = 2 * half;           // per-lane K sub-offset

  // Bias folded into the accumulator: C[M][N] = b[N], N = 16j + lm.
  v8f acc[4];
#pragma unroll
  for (int j = 0; j < 4; ++j) {
    float bj = b[16 * j + lm];
#pragma unroll
    for (int r = 0; r < 8; ++r) acc[j][r] = bj;
  }

  const float* nrow = nf  + (size_t)arow * kNodeDim + koff;
  const float* grow = agg + (size_t)arow * kAggDim  + koff;
  const float* wrow = W   + (size_t)lm   * kInDim   + koff;  // row lm of W; cols +16*j*kInDim

  // --- K = 0..63 : node-feature half of X ---
  for (int k = 0; k < kNodeDim; k += 4) {
    v2f a = *(const v2f*)(nrow + k);
#pragma unroll
    for (int j = 0; j < 4; ++j) {
      v2f bb = *(const v2f*)(wrow + (size_t)(16 * j) * kInDim + k);
      acc[j] = __builtin_amdgcn_wmma_f32_16x16x4_f32(
          false, a, false, bb, (short)0, acc[j], false, false);
    }
  }

  // --- K = 64..159 : aggregated-mailbox half of X ---
  for (int k = 0; k < kAggDim; k += 4) {
    v2f a = *(const v2f*)(grow + k);
#pragma unroll
    for (int j = 0; j < 4; ++j) {
      v2f bb = *(const v2f*)(wrow + (size_t)(16 * j) * kInDim + kNodeDim + k);
      acc[j] = __builtin_amdgcn_wmma_f32_16x16x4_f32(
          false, a, false, bb, (short)0, acc[j], false, false);
    }
  }

  // ReLU + store (streaming: output is write-once).
#pragma unroll
  for (int j = 0; j < 4; ++j) {
    const int col = 16 * j + lm;
#pragma unroll
    for (int r = 0; r < 8; ++r) {
      const int orow = rowbase + r + 8 * half;
      float v = acc[j][r];
      v = v > 0.f ? v : 0.f;
      __builtin_nontemporal_store(v, out + (size_t)orow * kOut + col);
    }
  }
}

// ---------------------------------------------------------------------------
extern "C" void kernel_launch(void* const* d_in, const int* in_sizes, int n_in,
                              void* d_out, int out_size, void* d_ws, size_t ws_size,
                              hipStream_t stream) {
  const float* nf  = (const float*)d_in[0];   // [50000, 64]
  const float* ef  = (const float*)d_in[1];   // [800000, 32]
  const int*   src = (const int*)d_in[2];     // [800000]
  const int*   dst = (const int*)d_in[3];     // [800000]
  const float* W   = (const float*)d_in[4];   // [64, 160]
  const float* b   = (const float*)d_in[5];   // [64]
  float* out = (float*)d_out;                 // [50000, 64]
  float* agg = (float*)d_ws;                  // [50000, 96] scratch mailboxes

  // Phase 0: zero mailboxes (float4 stores).
  const int n4 = kNodes * kAggDim / 4;        // 1,200,000
  zero_ws_kernel<<<(n4 + 255) / 256, 256, 0, stream>>>((float4*)agg, n4);

  // Phase 1: edge scatter, 32 lanes per edge.
  const long long scatterThreads = (long long)kEdges * 32;   // 25.6M
  edge_scatter_kernel<<<(int)(scatterThreads / 256), 256, 0, stream>>>(
      nf, ef, src, dst, agg);

  // Phase 2: WMMA node update, one wave per 16-row tile.
  const int updThreads = kRowTiles * 32;      // 100,000
  node_update_wmma_kernel<<<(updThreads + 255) / 256, 256, 0, stream>>>(
      nf, agg, W, b, out);
}